// Diffusion_Model_UQ_70677981823604
// MI455X (gfx1250) — compile-verified
//
#include <hip/hip_runtime.h>
#include <hip/hip_bf16.h>
#include <math.h>

#define N_NODES 20000
#define N_EDGES 320000
#define BATCH   8
#define T_IN    12
#define H_VEL   32
#define H_PROB  64

typedef __attribute__((ext_vector_type(2))) float v2f;
typedef __attribute__((ext_vector_type(8))) float v8f;

// ---------------------------------------------------------------------------
// WMMA helpers: V_WMMA_F32_16X16X4_F32, fp32 in/out, K=4 per issue.
// A 16x4: lanes 0-15 -> M=lane, VGPR0=K+0,VGPR1=K+1; lanes 16-31 -> K+2,K+3.
// B 4x16: lanes 0-15 -> N=lane, VGPR0=K+0,VGPR1=K+1; lanes 16-31 -> K+2,K+3.
// C/D 16x16: VGPR r -> M=r (lanes 0-15), M=r+8 (lanes 16-31), N=lane%16.
// ---------------------------------------------------------------------------
__device__ __forceinline__ v8f wmma_k4(v2f a, v2f b, v8f c) {
  return __builtin_amdgcn_wmma_f32_16x16x4_f32(false, a, false, b,
                                               (short)0, c, false, false);
}

// sum across each 16-lane half of the wave (halves stay independent)
__device__ __forceinline__ float hsum16(float x) {
  x += __shfl_xor(x, 1, 32);
  x += __shfl_xor(x, 2, 32);
  x += __shfl_xor(x, 4, 32);
  x += __shfl_xor(x, 8, 32);
  return x;
}

// A fragment: 16 rows of X[row][0..11], K-chunk kc in {0,4,8}
__device__ __forceinline__ v2f load_afrag(const float* __restrict__ X,
                                          size_t rowBase, int kc, int lane) {
  size_t row = rowBase + (size_t)(lane & 15);
  int k = kc + ((lane & 16) ? 2 : 0);
  const float* p = X + row * T_IN + k;
  v2f r; r.x = p[0]; r.y = p[1];
  return r;
}

// B fragment from weight W[H][12]: B[k][n] = W[n][k], N-tile nt, K-chunk kc
__device__ __forceinline__ v2f load_bfrag(const float* __restrict__ W,
                                          int nt, int kc, int lane) {
  int n = nt * 16 + (lane & 15);
  int k = kc + ((lane & 16) ? 2 : 0);
  const float* p = W + n * T_IN + k;
  v2f r; r.x = p[0]; r.y = p[1];
  return r;
}

// ---------------------------------------------------------------------------
// One velocity branch for 16 rows: sigmoid(LNReLU(x@w1^T + b1) @ w2^T + b2)
// out[r] is valid across all 16 lanes of the half holding that row.
// ---------------------------------------------------------------------------
__device__ __forceinline__ void branch16(const float* __restrict__ X, size_t rowBase,
                                         const float* __restrict__ w1,
                                         const float* __restrict__ b1,
                                         const float* __restrict__ g,
                                         const float* __restrict__ bn,
                                         const float* __restrict__ w2, float b2,
                                         int lane, float out[8]) {
  v2f a0 = load_afrag(X, rowBase, 0, lane);
  v2f a1 = load_afrag(X, rowBase, 4, lane);
  v2f a2 = load_afrag(X, rowBase, 8, lane);

  v8f acc0 = {}; v8f acc1 = {};
  acc0 = wmma_k4(a0, load_bfrag(w1, 0, 0, lane), acc0);
  acc0 = wmma_k4(a1, load_bfrag(w1, 0, 4, lane), acc0);
  acc0 = wmma_k4(a2, load_bfrag(w1, 0, 8, lane), acc0);
  acc1 = wmma_k4(a0, load_bfrag(w1, 1, 0, lane), acc1);
  acc1 = wmma_k4(a1, load_bfrag(w1, 1, 4, lane), acc1);
  acc1 = wmma_k4(a2, load_bfrag(w1, 1, 8, lane), acc1);

  int n0 = lane & 15;            // hidden index covered by acc0 in this lane
  int n1 = 16 + (lane & 15);     // hidden index covered by acc1
  float bias0 = b1[n0], bias1 = b1[n1];
  float g0 = g[n0], g1 = g[n1];
  float be0 = bn[n0], be1 = bn[n1];
  float wa = w2[n0], wb = w2[n1];

#pragma unroll
  for (int r = 0; r < 8; ++r) {
    float h0 = acc0[r] + bias0;
    float h1 = acc1[r] + bias1;
    float mu = hsum16(h0 + h1) * (1.0f / 32.0f);
    float d0 = h0 - mu, d1 = h1 - mu;
    float var = hsum16(d0 * d0 + d1 * d1) * (1.0f / 32.0f);
    float inv = rsqrtf(var + 1e-5f);
    float r0 = fmaxf(d0 * inv * g0 + be0, 0.0f);
    float r1 = fmaxf(d1 * inv * g1 + be1, 0.0f);
    float dot = hsum16(r0 * wa + r1 * wb) + b2;
    out[r] = 1.0f / (1.0f + expf(-dot));
  }
}

// ---------------------------------------------------------------------------
// Kernel V: velocity v[E*B] = softplus([xu,xd] @ v3_w^T + v3_b)
// wave handles 16 (e,b) rows; 12 WMMA per wave.
// ---------------------------------------------------------------------------
__global__ void velocity_kernel(const float* __restrict__ up,
                                const float* __restrict__ dn,
                                const float* __restrict__ v11w, const float* __restrict__ v11b,
                                const float* __restrict__ ln11g, const float* __restrict__ ln11b,
                                const float* __restrict__ v12w, const float* __restrict__ v12b,
                                const float* __restrict__ v21w, const float* __restrict__ v21b,
                                const float* __restrict__ ln21g, const float* __restrict__ ln21b,
                                const float* __restrict__ v22w, const float* __restrict__ v22b,
                                const float* __restrict__ v3w, const float* __restrict__ v3b,
                                float* __restrict__ v_out) {
  int lane = threadIdx.x & 31;
  size_t wave = (size_t)blockIdx.x * (blockDim.x >> 5) + (threadIdx.x >> 5);
  size_t rowBase = wave * 16;

  float xu[8], xd[8];
  branch16(up, rowBase, v11w, v11b, ln11g, ln11b, v12w, v12b[0], lane, xu);
  branch16(dn, rowBase, v21w, v21b, ln21g, ln21b, v22w, v22b[0], lane, xd);

  float w3x = v3w[0], w3y = v3w[1], b3 = v3b[0];
#pragma unroll
  for (int r = 0; r < 8; ++r) {
    float t = w3x * xu[r] + w3y * xd[r] + b3;
    float v = log1pf(expf(t));            // softplus, matches log1p(exp(t))
    if ((lane & 15) == r) {               // lane r writes row r, lane 16+r row r+8
      size_t row = rowBase + (size_t)r + ((lane & 16) ? 8 : 0);
      v_out[row] = v;
    }
  }
}

// ---------------------------------------------------------------------------
// Kernel Z: z[N*B][64] = feature @ fc_w^T  (12 WMMA per wave, 16 rows)
// ---------------------------------------------------------------------------
__global__ void z_kernel(const float* __restrict__ feature,
                         const float* __restrict__ fcw,
                         float* __restrict__ z) {
  int lane = threadIdx.x & 31;
  size_t wave = (size_t)blockIdx.x * (blockDim.x >> 5) + (threadIdx.x >> 5);
  size_t rowBase = wave * 16;

  v2f a0 = load_afrag(feature, rowBase, 0, lane);
  v2f a1 = load_afrag(feature, rowBase, 4, lane);
  v2f a2 = load_afrag(feature, rowBase, 8, lane);

  size_t rowHalf = rowBase + ((lane & 16) ? 8 : 0);
#pragma unroll
  for (int nt = 0; nt < 4; ++nt) {
    v8f acc = {};
    acc = wmma_k4(a0, load_bfrag(fcw, nt, 0, lane), acc);
    acc = wmma_k4(a1, load_bfrag(fcw, nt, 4, lane), acc);
    acc = wmma_k4(a2, load_bfrag(fcw, nt, 8, lane), acc);
    int n = nt * 16 + (lane & 15);
#pragma unroll
    for (int r = 0; r < 8; ++r)
      z[(rowHalf + r) * H_PROB + n] = acc[r];   // lanes 0-15 contiguous: coalesced
  }
}

// ordered-uint encode/decode for float atomic max
__device__ __forceinline__ unsigned f2key(float f) {
  unsigned b = __float_as_uint(f);
  return (b & 0x80000000u) ? ~b : (b | 0x80000000u);
}
__device__ __forceinline__ float key2f(unsigned k) {
  return __uint_as_float((k & 0x80000000u) ? (k & 0x7FFFFFFFu) : ~k);
}

// ---------------------------------------------------------------------------
// Prep: gw[i] = ln2_g[i]*attn_w[i]; scal = {sum gw, sum ln2_b*attn_w}
// Single block; amortizes per-edge LN/dot algebra across 2.56M threads.
// ---------------------------------------------------------------------------
__global__ void prep_kernel(const float* __restrict__ attn_w,
                            const float* __restrict__ ln2g,
                            const float* __restrict__ ln2b,
                            float* __restrict__ gw_buf,
                            float* __restrict__ scal) {
  int i = threadIdx.x;
  if (i >= 129 && i < 132) gw_buf[i] = 0.0f;          // pad
  if (i < 129) gw_buf[i] = ln2g[i] * attn_w[i];
  __syncthreads();
  if (i == 0) {
    float sg = 0.0f, sb = 0.0f;
    for (int k = 0; k < 129; ++k) { sg += gw_buf[k]; sb += ln2b[k] * attn_w[k]; }
    scal[0] = sg;
    scal[1] = sb;
  }
}

// ---------------------------------------------------------------------------
// Kernel E: per (e,b): diffusion pred_edge, attention logit a, segment-max m
// LN(zc)@attn_w refactored: a = (Sxgw - mu*Sgw)/sigma + Sbw  (single pass,
// b128 loads from the L2-resident z table).
// ---------------------------------------------------------------------------
__global__ void edge_kernel(const float* __restrict__ feature,
                            const float* __restrict__ distance,
                            const int* __restrict__ src, const int* __restrict__ dst,
                            const float* __restrict__ alpha,
                            const float* __restrict__ z,
                            const float* __restrict__ v_arr,
                            const float* __restrict__ gw_buf,
                            const float* __restrict__ scal,
                            float* __restrict__ a_out,
                            float* __restrict__ pe_out,
                            unsigned* __restrict__ m_keys) {
  size_t idx = (size_t)blockIdx.x * blockDim.x + threadIdx.x;
  if (idx >= (size_t)N_EDGES * BATCH) return;
  int e = (int)(idx >> 3);
  int b = (int)(idx & 7);
  int s = src[e], d = dst[e];

  float v  = v_arr[idx];
  float Tt = distance[e] / (v + 1e-5f);
  float ti = rintf(Tt * (1.0f / 10.0f));        // round half-to-even like jnp.round
  if (ti < 0.0f)  ti = 0.0f;
  if (ti > 12.0f) ti = 11.0f;                   // strictly > T, as in reference
  float F = 1.0f / (1.0f + alpha[e] * Tt);
  int ni = (int)fmaxf(12.0f - ti, 1.0f);
  float gf = 1.0f - F;

  // source feature row: 48B stride -> 16B aligned, three b128 loads
  const float4* feat4 = (const float4*)(feature + ((size_t)s * BATCH + b) * T_IN);
  float4 f0 = feat4[0], f1 = feat4[1], f2 = feat4[2];
  float fr[12] = {f0.x, f0.y, f0.z, f0.w, f1.x, f1.y, f1.z, f1.w,
                  f2.x, f2.y, f2.z, f2.w};
  float hs = 0.0f;                               // Horner: sum fr[t]*gf^(ni-1-t)
#pragma unroll
  for (int t = 0; t < 12; ++t)
    if (t < ni) hs = hs * gf + fr[t];
  pe_out[idx] = F * hs;

  // attention logit: leaky_relu( LN([z_src, z_dst, v]) @ attn_w )
  const float4* zs  = (const float4*)(z + ((size_t)s * BATCH + b) * H_PROB);
  const float4* zd  = (const float4*)(z + ((size_t)d * BATCH + b) * H_PROB);
  const float4* gw4 = (const float4*)gw_buf;
  float Sx = 0.0f, Sxx = 0.0f, Sxgw = 0.0f;
#pragma unroll 4
  for (int i = 0; i < 16; ++i) {
    float4 x = zs[i]; float4 g = gw4[i];
    Sx   += (x.x + x.y) + (x.z + x.w);
    Sxx  += (x.x * x.x + x.y * x.y) + (x.z * x.z + x.w * x.w);
    Sxgw += (x.x * g.x + x.y * g.y) + (x.z * g.z + x.w * g.w);
  }
#pragma unroll 4
  for (int i = 0; i < 16; ++i) {
    float4 x = zd[i]; float4 g = gw4[16 + i];
    Sx   += (x.x + x.y) + (x.z + x.w);
    Sxx  += (x.x * x.x + x.y * x.y) + (x.z * x.z + x.w * x.w);
    Sxgw += (x.x * g.x + x.y * g.y) + (x.z * g.z + x.w * g.w);
  }
  Sx += v; Sxx += v * v; Sxgw += v * gw_buf[128];

  const float invC = 1.0f / 129.0f;
  float mu  = Sx * invC;
  float var = fmaxf(Sxx * invC - mu * mu, 0.0f);
  float inv = rsqrtf(var + 1e-5f);
  float a   = (Sxgw - mu * scal[0]) * inv + scal[1];
  a = (a > 0.0f) ? a : 0.01f * a;                // leaky_relu(0.01)
  a_out[idx] = a;
  atomicMax(&m_keys[(size_t)s * BATCH + b], f2key(a));
}

// Kernel S: ex = exp(a - m[src]) (in place over a); den[src] += ex
__global__ void softmax_kernel(const int* __restrict__ src,
                               const unsigned* __restrict__ m_keys,
                               float* __restrict__ a_ex,
                               float* __restrict__ den) {
  size_t idx = (size_t)blockIdx.x * blockDim.x + threadIdx.x;
  if (idx >= (size_t)N_EDGES * BATCH) return;
  int e = (int)(idx >> 3);
  int b = (int)(idx & 7);
  size_t srow = (size_t)src[e] * BATCH + b;
  float ex = expf(a_ex[idx] - key2f(m_keys[srow]));
  a_ex[idx] = ex;
  atomicAdd(&den[srow], ex);
}

// Kernel F: pred[dst] += (ex/den[src]) * pred_edge
__global__ void finalize_kernel(const int* __restrict__ src, const int* __restrict__ dst,
                                const float* __restrict__ ex, const float* __restrict__ den,
                                const float* __restrict__ pe,
                                float* __restrict__ pred) {
  size_t idx = (size_t)blockIdx.x * blockDim.x + threadIdx.x;
  if (idx >= (size_t)N_EDGES * BATCH) return;
  int e = (int)(idx >> 3);
  int b = (int)(idx & 7);
  float score = ex[idx] / den[(size_t)src[e] * BATCH + b];
  atomicAdd(&pred[(size_t)dst[e] * BATCH + b], score * pe[idx]);
}

// ---------------------------------------------------------------------------
extern "C" void kernel_launch(void* const* d_in, const int* in_sizes, int n_in,
                              void* d_out, int out_size, void* d_ws, size_t ws_size,
                              hipStream_t stream) {
  const float* upstream   = (const float*)d_in[0];
  const float* downstream = (const float*)d_in[1];
  const float* feature    = (const float*)d_in[2];
  const float* distance   = (const float*)d_in[3];
  const int*   src        = (const int*)d_in[4];
  const int*   dst        = (const int*)d_in[5];
  const float* fc_w       = (const float*)d_in[6];
  const float* attn_w     = (const float*)d_in[7];
  const float* ln2_g      = (const float*)d_in[8];
  const float* ln2_b      = (const float*)d_in[9];
  const float* v11_w      = (const float*)d_in[10];
  const float* v11_b      = (const float*)d_in[11];
  const float* ln11_g     = (const float*)d_in[12];
  const float* ln11_b     = (const float*)d_in[13];
  const float* v12_w      = (const float*)d_in[14];
  const float* v12_b      = (const float*)d_in[15];
  const float* v21_w      = (const float*)d_in[16];
  const float* v21_b      = (const float*)d_in[17];
  const float* ln21_g     = (const float*)d_in[18];
  const float* ln21_b     = (const float*)d_in[19];
  const float* v22_w      = (const float*)d_in[20];
  const float* v22_b      = (const float*)d_in[21];
  const float* v3_w       = (const float*)d_in[22];
  const float* v3_b       = (const float*)d_in[23];
  const float* alpha      = (const float*)d_in[24];

  const size_t ROWS_N = (size_t)N_NODES * BATCH;        // 160,000
  const size_t ROWS_E = (size_t)N_EDGES * BATCH;        // 2,560,000

  char* ws = (char*)d_ws;
  float*    z_buf  = (float*)ws;                         //  40,960,000 B
  float*    v_buf  = (float*)(ws + 40960000);            //  10,240,000 B
  float*    a_buf  = (float*)(ws + 51200000);            //  10,240,000 B (a, then ex)
  float*    p_buf  = (float*)(ws + 61440000);            //  10,240,000 B (pred_edge)
  unsigned* m_buf  = (unsigned*)(ws + 71680000);         //     640,000 B
  float*    d_buf  = (float*)(ws + 72320000);            //     640,000 B (den)
  float*    gw_buf = (float*)(ws + 72960000);            //         528 B (pad 1024)
  float*    s_buf  = (float*)(ws + 72961024);            //           8 B (scalars)

  hipMemsetAsync(m_buf, 0, ROWS_N * sizeof(unsigned), stream);  // key 0 < all real keys
  hipMemsetAsync(d_buf, 0, ROWS_N * sizeof(float), stream);
  hipMemsetAsync(d_out, 0, ROWS_N * sizeof(float), stream);

  prep_kernel<<<1, 256, 0, stream>>>(attn_w, ln2_g, ln2_b, gw_buf, s_buf);

  // 16 rows per wave, 8 waves per 256-thread block
  z_kernel<<<(int)(ROWS_N / 16 / 8), 256, 0, stream>>>(feature, fc_w, z_buf);

  velocity_kernel<<<(int)(ROWS_E / 16 / 8), 256, 0, stream>>>(
      upstream, downstream,
      v11_w, v11_b, ln11_g, ln11_b, v12_w, v12_b,
      v21_w, v21_b, ln21_g, ln21_b, v22_w, v22_b,
      v3_w, v3_b, v_buf);

  int ethreads = 256;
  int eblocks = (int)((ROWS_E + ethreads - 1) / ethreads);
  edge_kernel<<<eblocks, ethreads, 0, stream>>>(
      feature, distance, src, dst, alpha,
      z_buf, v_buf, gw_buf, s_buf, a_buf, p_buf, m_buf);

  softmax_kernel<<<eblocks, ethreads, 0, stream>>>(src, m_buf, a_buf, d_buf);

  finalize_kernel<<<eblocks, ethreads, 0, stream>>>(src, dst, a_buf, d_buf, p_buf,
                                                    (float*)d_out);
}